// GNN_51324859187584
// MI455X (gfx1250) — compile-verified
//
#include <hip/hip_runtime.h>
#include <hip/hip_bf16.h>

typedef __attribute__((ext_vector_type(2))) float v2f;
typedef __attribute__((ext_vector_type(8))) float v8f;

#define H 128
#define LDS_STRIDE 260  // 16x256 A panel, +4 pad: bank-conflict-free, 8B aligned b64 loads

// ---------------------------------------------------------------------------
// Prologue: out[row] = emb[nid[row]] + x[row] @ W + b      (x: [N,F], W: [F,H])
// ---------------------------------------------------------------------------
template<int F>
__global__ __launch_bounds__(128)
void prologue_kernel(const float* __restrict__ x, const float* __restrict__ emb,
                     const int* __restrict__ nid, const float* __restrict__ W,
                     const float* __restrict__ b, float* __restrict__ out, int N) {
    __shared__ float Wl[F * H];
    __shared__ float bl[H];
    for (int i = threadIdx.x; i < F * H; i += 128) Wl[i] = W[i];
    if (threadIdx.x < H) bl[threadIdx.x] = b[threadIdx.x];
    __syncthreads();
    const int ROWS = 8;
    int row0 = blockIdx.x * ROWS;
    int h = threadIdx.x;
    for (int r = 0; r < ROWS; ++r) {
        int row = row0 + r;
        if (row >= N) break;
        float acc = emb[(size_t)nid[row] * H + h] + bl[h];
#pragma unroll
        for (int f = 0; f < F; ++f)
            acc = fmaf(x[(size_t)row * F + f], Wl[f * H + h], acc);
        out[(size_t)row * H + h] = acc;
    }
}

// ---------------------------------------------------------------------------
// Scatter-add: one wave32 per edge. agg[dst] += xsrc[src]; cnt[dst] += 1.
// ---------------------------------------------------------------------------
__global__ __launch_bounds__(256)
void scatter_kernel(const float* __restrict__ xsrc, const int* __restrict__ src,
                    const int* __restrict__ dst, float* __restrict__ agg,
                    float* __restrict__ cnt, int nedges) {
    int warp = (int)((blockIdx.x * (size_t)blockDim.x + threadIdx.x) >> 5);
    int lane = threadIdx.x & 31;
    if (warp >= nedges) return;
    int s = src[warp];
    int d = dst[warp];
    const float4* xs = (const float4*)(xsrc + (size_t)s * H);
    float4 v = xs[lane];                       // 32 lanes x float4 = 128 floats
    float* ap = agg + (size_t)d * H + lane * 4;
    atomicAdd(ap + 0, v.x);
    atomicAdd(ap + 1, v.y);
    atomicAdd(ap + 2, v.z);
    atomicAdd(ap + 3, v.w);
    if (lane == 0) atomicAdd(cnt + d, 1.0f);
}

// ---------------------------------------------------------------------------
// Fused SAGE GEMM via V_WMMA_F32_16X16X4_F32:
//   out = [agg/max(cnt,1) | xd] (16x256) @ [Wmsg ; Wself] (256x128) + bias
// Block: 128 threads = 4 waves, 16 output rows; wave w -> columns [32w,32w+32).
// ---------------------------------------------------------------------------
__global__ __launch_bounds__(128)
void sage_wmma_kernel(const float* __restrict__ agg, const float* __restrict__ cnt,
                      const float* __restrict__ xd,
                      const float* __restrict__ Wmsg, const float* __restrict__ Wself,
                      const float* __restrict__ bias, float* __restrict__ out,
                      int N, int do_relu) {
    __shared__ float A[16 * LDS_STRIDE];
    int tid  = threadIdx.x;
    int lane = tid & 31;
    int wave = tid >> 5;
    int row0 = blockIdx.x * 16;

    // Stage normalized A panel into LDS: cols [0,128)=agg/cnt, [128,256)=self.
    for (int idx = tid; idx < 16 * 256; idx += 128) {
        int r = idx >> 8;
        int k = idx & 255;
        int row = row0 + r;
        float v = 0.0f;
        if (row < N) {
            if (k < H) {
                float c = cnt[row];
                c = c > 1.0f ? c : 1.0f;
                v = agg[(size_t)row * H + k] / c;
            } else {
                v = xd[(size_t)row * H + (k - H)];
            }
        }
        A[r * LDS_STRIDE + k] = v;
    }
    __syncthreads();

    // A frag (16x4 f32): lanes 0-15 hold M=lane K={k,k+1}; lanes 16-31 K={k+2,k+3}.
    // B frag (4x16 f32): VGPR0/1 hold K/K+1 (low lanes), K+2/K+3 (high lanes), N=lane&15.
    int m    = lane & 15;
    int koff = (lane >> 4) << 1;
    int col0 = wave * 32 + m;
    v8f c0 = {0.f, 0.f, 0.f, 0.f, 0.f, 0.f, 0.f, 0.f};
    v8f c1 = {0.f, 0.f, 0.f, 0.f, 0.f, 0.f, 0.f, 0.f};

    for (int half = 0; half < 2; ++half) {
        const float* __restrict__ W = half ? Wself : Wmsg;
        const float* Arow = &A[m * LDS_STRIDE + half * H];
#pragma unroll 4
        for (int kb = 0; kb < H; kb += 4) {
            int kk = kb + koff;
            v2f a = *(const v2f*)(Arow + kk);               // ds_load_b64
            v2f b0, b1;
            b0.x = W[(size_t)kk * H + col0];
            b0.y = W[(size_t)(kk + 1) * H + col0];
            b1.x = W[(size_t)kk * H + col0 + 16];
            b1.y = W[(size_t)(kk + 1) * H + col0 + 16];
            c0 = __builtin_amdgcn_wmma_f32_16x16x4_f32(false, a, false, b0,
                                                       (short)0, c0, false, false);
            c1 = __builtin_amdgcn_wmma_f32_16x16x4_f32(false, a, false, b1,
                                                       (short)0, c1, false, false);
        }
    }

    // Epilogue: C/D layout — VGPR i: lanes 0-15 -> M=i, lanes 16-31 -> M=i+8.
    float bi0 = bias[col0];
    float bi1 = bias[col0 + 16];
    int mbase = (lane >> 4) << 3;
#pragma unroll
    for (int i = 0; i < 8; ++i) {
        int row = row0 + mbase + i;
        if (row < N) {
            float v0 = c0[i] + bi0;
            float v1 = c1[i] + bi1;
            if (do_relu) { v0 = fmaxf(v0, 0.0f); v1 = fmaxf(v1, 0.0f); }
            out[(size_t)row * H + col0]      = v0;
            out[(size_t)row * H + col0 + 16] = v1;
        }
    }
}

// ---------------------------------------------------------------------------
// Classifier: pred[e] = dot(xu2[el_src[e]], xa2[el_dst[e]]). One wave per pair.
// ---------------------------------------------------------------------------
__global__ __launch_bounds__(256)
void edge_dot_kernel(const float* __restrict__ xu2, const float* __restrict__ xa2,
                     const int* __restrict__ esrc, const int* __restrict__ edst,
                     float* __restrict__ outp, int nedges) {
    int warp = (int)((blockIdx.x * (size_t)blockDim.x + threadIdx.x) >> 5);
    int lane = threadIdx.x & 31;
    if (warp >= nedges) return;
    const float4* u = (const float4*)(xu2 + (size_t)esrc[warp] * H);
    const float4* a = (const float4*)(xa2 + (size_t)edst[warp] * H);
    float4 uv = u[lane];
    float4 av = a[lane];
    float s = uv.x * av.x + uv.y * av.y + uv.z * av.z + uv.w * av.w;
#pragma unroll
    for (int off = 16; off >= 1; off >>= 1)
        s += __shfl_xor(s, off, 32);
    if (lane == 0) outp[warp] = s;
}

// ---------------------------------------------------------------------------
extern "C" void kernel_launch(void* const* d_in, const int* in_sizes, int n_in,
                              void* d_out, int out_size, void* d_ws, size_t ws_size,
                              hipStream_t stream) {
    (void)n_in; (void)out_size; (void)ws_size;

    const float* user_x     = (const float*)d_in[0];
    const float* app_x      = (const float*)d_in[1];
    const float* user_emb   = (const float*)d_in[2];
    const float* app_emb    = (const float*)d_in[3];
    const float* user_lin_W = (const float*)d_in[4];
    const float* user_lin_b = (const float*)d_in[5];
    const float* app_lin_W  = (const float*)d_in[6];
    const float* app_lin_b  = (const float*)d_in[7];
    const float* W_msg_ua1  = (const float*)d_in[8];
    const float* W_self_ua1 = (const float*)d_in[9];
    const float* b_ua1      = (const float*)d_in[10];
    const float* W_msg_au1  = (const float*)d_in[11];
    const float* W_self_au1 = (const float*)d_in[12];
    const float* b_au1      = (const float*)d_in[13];
    const float* W_msg_ua2  = (const float*)d_in[14];
    const float* W_self_ua2 = (const float*)d_in[15];
    const float* b_ua2      = (const float*)d_in[16];
    const float* W_msg_au2  = (const float*)d_in[17];
    const float* W_self_au2 = (const float*)d_in[18];
    const float* b_au2      = (const float*)d_in[19];
    const int* user_n_id    = (const int*)d_in[20];
    const int* app_n_id     = (const int*)d_in[21];
    const int* e_src_u2a    = (const int*)d_in[22];
    const int* e_dst_u2a    = (const int*)d_in[23];
    const int* e_src_a2u    = (const int*)d_in[24];
    const int* e_dst_a2u    = (const int*)d_in[25];
    const int* el_src       = (const int*)d_in[26];
    const int* el_dst       = (const int*)d_in[27];

    const int NU  = in_sizes[20];
    const int NA  = in_sizes[21];
    const int E_  = in_sizes[22];
    const int EL_ = in_sizes[26];

    // Workspace carve-up (floats). xu/xa are recycled as xu2/xa2.
    float* w = (float*)d_ws;
    size_t off = 0;
    float* xu    = w + off; off += (size_t)NU * H;
    float* xa    = w + off; off += (size_t)NA * H;
    float* xu1   = w + off; off += (size_t)NU * H;
    float* xa1   = w + off; off += (size_t)NA * H;
    float* agg_u = w + off; off += (size_t)NU * H;
    float* agg_a = w + off; off += (size_t)NA * H;
    float* cnt_u = w + off; off += (size_t)NU;
    float* cnt_a = w + off; off += (size_t)NA;

    // Prologue
    prologue_kernel<32><<<(NU + 7) / 8, 128, 0, stream>>>(
        user_x, user_emb, user_n_id, user_lin_W, user_lin_b, xu, NU);
    prologue_kernel<64><<<(NA + 7) / 8, 128, 0, stream>>>(
        app_x, app_emb, app_n_id, app_lin_W, app_lin_b, xa, NA);

    auto scatter = [&](const float* xsrc, const int* src, const int* dst,
                       float* agg, float* cnt, int n_dst) {
        hipMemsetAsync(agg, 0, (size_t)n_dst * H * sizeof(float), stream);
        hipMemsetAsync(cnt, 0, (size_t)n_dst * sizeof(float), stream);
        size_t threads = (size_t)E_ * 32;
        scatter_kernel<<<(unsigned)((threads + 255) / 256), 256, 0, stream>>>(
            xsrc, src, dst, agg, cnt, E_);
    };

    // Layer 1 (ReLU)
    scatter(xu, e_src_u2a, e_dst_u2a, agg_a, cnt_a, NA);
    sage_wmma_kernel<<<(NA + 15) / 16, 128, 0, stream>>>(
        agg_a, cnt_a, xa, W_msg_ua1, W_self_ua1, b_ua1, xa1, NA, 1);
    scatter(xa, e_src_a2u, e_dst_a2u, agg_u, cnt_u, NU);
    sage_wmma_kernel<<<(NU + 15) / 16, 128, 0, stream>>>(
        agg_u, cnt_u, xu, W_msg_au1, W_self_au1, b_au1, xu1, NU, 1);

    // Layer 2 (no activation); xu/xa recycled as xu2/xa2
    scatter(xu1, e_src_u2a, e_dst_u2a, agg_a, cnt_a, NA);
    sage_wmma_kernel<<<(NA + 15) / 16, 128, 0, stream>>>(
        agg_a, cnt_a, xa1, W_msg_ua2, W_self_ua2, b_ua2, xa, NA, 0);
    scatter(xa1, e_src_a2u, e_dst_a2u, agg_u, cnt_u, NU);
    sage_wmma_kernel<<<(NU + 15) / 16, 128, 0, stream>>>(
        agg_u, cnt_u, xu1, W_msg_au2, W_self_au2, b_au2, xu, NU, 0);

    // Classifier
    size_t dthreads = (size_t)EL_ * 32;
    edge_dot_kernel<<<(unsigned)((dthreads + 255) / 256), 256, 0, stream>>>(
        xu, xa, el_src, el_dst, (float*)d_out, EL_);
}